// MyHeroGraphConvTest_48275432407743
// MI455X (gfx1250) — compile-verified
//
#include <hip/hip_runtime.h>

#define NN 50000
#define NE 800000
#define DD 128

typedef __attribute__((ext_vector_type(2))) float v2f;
typedef __attribute__((ext_vector_type(8))) float v8f;

// ---------------------------------------------------------------- zero
__global__ void hgc_zero_kernel(float* __restrict__ p, int n) {
  int i = blockIdx.x * blockDim.x + threadIdx.x;
  if (i < n) p[i] = 0.0f;
}

// ---------------------------------------------------------------- copy
__global__ void hgc_copy_kernel(const float* __restrict__ s, float* __restrict__ d, int n) {
  int i = blockIdx.x * blockDim.x + threadIdx.x;
  if (i < n) d[i] = s[i];
}

// ---------------------------------------------------------------- degrees
__global__ void hgc_degree_kernel(const int* __restrict__ src,
                                  const int* __restrict__ dst,
                                  float* __restrict__ deg_src,
                                  float* __restrict__ deg_dst, int e) {
  int i = blockIdx.x * blockDim.x + threadIdx.x;
  if (i < e) {
    atomicAdd(&deg_src[src[i]], 1.0f);
    atomicAdd(&deg_dst[dst[i]], 1.0f);
  }
}

// ---------------------------------------------------------------- scatter-add
// 256 threads = 2 edges per block; thread t handles feature column t&127.
__global__ __launch_bounds__(256)
void hgc_scatter_kernel(const float* __restrict__ feat,
                        const int* __restrict__ src,
                        const int* __restrict__ dst,
                        const float* __restrict__ deg_src,
                        float* __restrict__ agg, int e) {
  int edge = blockIdx.x * 2 + (threadIdx.x >> 7);
  int t = threadIdx.x & 127;
  if (edge < e) {
    int s = src[edge];
    int d = dst[edge];
    float w = rsqrtf(fmaxf(deg_src[s], 1.0f));
    atomicAdd(&agg[d * DD + t], feat[s * DD + t] * w);
  }
}

// ---------------------------------------------------------------- WMMA GEMM
// out[m, n] = sum_k (agg[m,k] * rsqrt(deg_dst[m])) * W[k,n] + bias[n]
//             (+ previous out[m,n] if accumulate)
// One block (8 waves) per 16-row strip of A; wave w owns columns 16w..16w+15.
__global__ __launch_bounds__(256)
void hgc_gemm_wmma_kernel(const float* __restrict__ agg,
                          const float* __restrict__ deg_dst,
                          const float* __restrict__ Wm,    // [128,128] row-major
                          const float* __restrict__ bias,  // [128]
                          float* __restrict__ out,
                          int accumulate) {
  const int wave   = threadIdx.x >> 5;       // 0..7 -> column tile
  const int lane   = threadIdx.x & 31;
  const int hi     = lane >> 4;              // 0: lanes 0-15, 1: lanes 16-31
  const int tile_m = blockIdx.x * 16;
  const int col    = wave * 16 + (lane & 15);

  // Stage the 16x128 A strip in LDS with deg_dst row scaling applied.
  __shared__ float As[16 * DD];
  for (int i = threadIdx.x; i < 16 * DD; i += 256) {
    int r = i >> 7;        // row within strip
    int c = i & 127;       // k
    float sc = rsqrtf(fmaxf(deg_dst[tile_m + r], 1.0f));
    As[i] = agg[(tile_m + r) * DD + c] * sc;
  }
  __syncthreads();

  // C/D 16x16 f32 layout: lane holds column col; 8 VGPRs are rows
  // m = tile_m + hi*8 + v. Fold bias (and optional previous output) into C.
  v8f acc;
#pragma unroll
  for (int v = 0; v < 8; ++v) {
    float x = bias[col];
    if (accumulate) x += out[(tile_m + hi * 8 + v) * DD + col];
    acc[v] = x;
  }

  const int am = lane & 15;  // A row within tile for this lane
#pragma unroll 4
  for (int k = 0; k < DD; k += 4) {
    // A 16x4 f32 fragment: lanes 0-15 carry K=k,k+1; lanes 16-31 carry K=k+2,k+3
    v2f a;
    a[0] = As[am * DD + k + hi * 2];
    a[1] = As[am * DD + k + hi * 2 + 1];
    // B 4x16 f32 fragment: VGPR0 = rows K=k (lanes 0-15) / K=k+2 (lanes 16-31),
    //                      VGPR1 = rows K=k+1 / K=k+3
    v2f b;
    b[0] = Wm[(k + hi * 2) * DD + col];
    b[1] = Wm[(k + hi * 2 + 1) * DD + col];
    acc = __builtin_amdgcn_wmma_f32_16x16x4_f32(
        /*neg_a=*/false, a, /*neg_b=*/false, b,
        /*c_mod=*/(short)0, acc, /*reuse_a=*/false, /*reuse_b=*/false);
  }

#pragma unroll
  for (int v = 0; v < 8; ++v) {
    out[(tile_m + hi * 8 + v) * DD + col] = acc[v];
  }
}

// ---------------------------------------------------------------- launcher
extern "C" void kernel_launch(void* const* d_in, const int* in_sizes, int n_in,
                              void* d_out, int out_size, void* d_ws, size_t ws_size,
                              hipStream_t stream) {
  (void)in_sizes; (void)n_in; (void)out_size; (void)ws_size;

  // Relation order: i2u_d1, i2u_d2, u2i_d1, u2i_d2
  const float* feat[4] = {(const float*)d_in[2], (const float*)d_in[3],
                          (const float*)d_in[0], (const float*)d_in[1]};
  const float* Wm[4]   = {(const float*)d_in[4], (const float*)d_in[6],
                          (const float*)d_in[8], (const float*)d_in[10]};
  const float* bm[4]   = {(const float*)d_in[5], (const float*)d_in[7],
                          (const float*)d_in[9], (const float*)d_in[11]};
  const int* srcs[4]   = {(const int*)d_in[12], (const int*)d_in[14],
                          (const int*)d_in[16], (const int*)d_in[18]};
  const int* dsts[4]   = {(const int*)d_in[13], (const int*)d_in[15],
                          (const int*)d_in[17], (const int*)d_in[19]};

  float* out = (float*)d_out;
  // Output slots: [user, user, out_d1, out_d2], each N*D floats.
  float* outs[4] = {out, out, out + 2 * (size_t)NN * DD, out + 3 * (size_t)NN * DD};
  const int accflag[4] = {0, 1, 0, 0};

  // Workspace: deg_src[N] | deg_dst[N] | agg[N*D]  (contiguous, zeroed together)
  float* ws      = (float*)d_ws;
  float* deg_src = ws;
  float* deg_dst = ws + NN;
  float* agg     = ws + 2 * NN;
  const int zcount = 2 * NN + NN * DD;

  for (int r = 0; r < 4; ++r) {
    hgc_zero_kernel<<<(zcount + 255) / 256, 256, 0, stream>>>(deg_src, zcount);
    hgc_degree_kernel<<<(NE + 255) / 256, 256, 0, stream>>>(
        srcs[r], dsts[r], deg_src, deg_dst, NE);
    hgc_scatter_kernel<<<(NE + 1) / 2, 256, 0, stream>>>(
        feat[r], srcs[r], dsts[r], deg_src, agg, NE);
    hgc_gemm_wmma_kernel<<<NN / 16, 256, 0, stream>>>(
        agg, deg_dst, Wm[r], bm[r], outs[r], accflag[r]);
  }

  // user slot duplicated (stack([user, user, ...]))
  hgc_copy_kernel<<<((NN * DD) + 255) / 256, 256, 0, stream>>>(
      out, out + (size_t)NN * DD, NN * DD);
}